// RotationInvariantEncoder_44641890074843
// MI455X (gfx1250) — compile-verified
//
#include <hip/hip_runtime.h>
#include <hip/hip_bf16.h>

typedef __attribute__((ext_vector_type(2))) float v2f;
typedef __attribute__((ext_vector_type(8))) float v8f;

#define NPTS   1048576
#define NCLUST 50000
#define F1     128
#define F2     64
#define FO     64

// D = A(16x4 f32) x B(4x16 f32) + C(16x16 f32), full-precision CDNA5 WMMA
__device__ __forceinline__ v8f wmma_f32_k4(v2f a, v2f b, v8f c) {
  return __builtin_amdgcn_wmma_f32_16x16x4_f32(
      /*neg_a=*/false, a, /*neg_b=*/false, b,
      /*c_mod=*/(short)0, c, /*reuse_a=*/false, /*reuse_b=*/false);
}

__device__ __forceinline__ float relu_f(float v) { return v > 0.f ? v : 0.f; }

// ---------------------------------------------------------------------------
// Kernel 1: per-point fc1 (4->64) + fc2 (64->128) + segment-max into mx[C,128]
// One wave handles 16 points. 8 waves / block -> 128 points / block.
// ---------------------------------------------------------------------------
__global__ __launch_bounds__(256) void k_point_mlp(
    const float* __restrict__ inv_feats,   // [N,4]
    const int*   __restrict__ cluster,     // [N]
    const float* __restrict__ W1,          // [64,4]
    const float* __restrict__ b1,          // [64]
    const float* __restrict__ W2,          // [128,64]
    const float* __restrict__ b2,          // [128]
    float* __restrict__ mx)                // [C,128], pre-zeroed
{
  __shared__ float sfc1[8][16][F2 + 2];    // wave-private fc1 staging, padded
  const int lane = threadIdx.x & 31;
  const int wave = threadIdx.x >> 5;
  const int half = lane >> 4;              // 0: lanes 0-15, 1: lanes 16-31
  const int lm   = lane & 15;
  const int wave_pt = blockIdx.x * 128 + wave * 16;

  // ---- fc1: A = inv_feats[16 pts, 4] (single K=4 step), B tiles = W1^T
  v2f a1 = *(const v2f*)(inv_feats + (wave_pt + lm) * 4 + half * 2);
  #pragma unroll
  for (int j = 0; j < 4; ++j) {            // 4 output tiles of 16 feats
    v2f b = *(const v2f*)(W1 + (16 * j + lm) * 4 + half * 2);
    float cb = b1[16 * j + lm];
    v8f acc = {cb, cb, cb, cb, cb, cb, cb, cb};
    acc = wmma_f32_k4(a1, b, acc);
    #pragma unroll
    for (int r = 0; r < 8; ++r)
      sfc1[wave][r + half * 8][16 * j + lm] = relu_f(acc[r]);
  }

  // cluster ids for the 8 row-slots this lane owns in C/D layout
  int cid[8];
  #pragma unroll
  for (int r = 0; r < 8; ++r)
    cid[r] = cluster[wave_pt + r + half * 8];

  // ---- fc2: [16,64] x [64,128] = 8 tiles x 16 WMMA(K=4) chain, then
  //      relu + atomic segment-max (uint compare valid: values >= 0)
  #pragma unroll
  for (int j = 0; j < 8; ++j) {
    float cb = b2[16 * j + lm];
    v8f acc = {cb, cb, cb, cb, cb, cb, cb, cb};
    #pragma unroll
    for (int kk = 0; kk < 16; ++kk) {
      int k0 = kk * 4 + half * 2;
      v2f a = *(const v2f*)&sfc1[wave][lm][k0];
      v2f b = *(const v2f*)(W2 + (16 * j + lm) * F2 + k0);
      acc = wmma_f32_k4(a, b, acc);
    }
    #pragma unroll
    for (int r = 0; r < 8; ++r) {
      float v = relu_f(acc[r]);
      atomicMax((unsigned int*)(mx + (size_t)cid[r] * F1 + 16 * j + lm),
                __float_as_uint(v));
    }
  }
}

// ---------------------------------------------------------------------------
// Kernel 2: per-cluster fc3 (128->64) + fc4 (64->64) -> d_out[0 : C*64]
// One wave handles 16 clusters (C = 3125 * 16). Wave-uniform tail guard.
// ---------------------------------------------------------------------------
__global__ __launch_bounds__(256) void k_cluster_mlp(
    const float* __restrict__ mx,          // [C,128] (>= 0)
    const float* __restrict__ W3,          // [64,128]
    const float* __restrict__ b3,          // [64]
    const float* __restrict__ W4,          // [64,64]
    const float* __restrict__ b4,          // [64]
    float* __restrict__ fc4_out)           // [C,64]
{
  __shared__ float sfc3[8][16][F2 + 2];
  const int lane = threadIdx.x & 31;
  const int wave = threadIdx.x >> 5;
  const int half = lane >> 4;
  const int lm   = lane & 15;
  const int wc = (blockIdx.x * 8 + wave) * 16;
  if (wc >= NCLUST) return;                // uniform per wave; EXEC stays full

  // ---- fc3: [16,128] x [128,64]
  #pragma unroll
  for (int j = 0; j < 4; ++j) {
    float cb = b3[16 * j + lm];
    v8f acc = {cb, cb, cb, cb, cb, cb, cb, cb};
    #pragma unroll
    for (int kk = 0; kk < 32; ++kk) {
      int k0 = kk * 4 + half * 2;
      v2f a = *(const v2f*)(mx + (size_t)(wc + lm) * F1 + k0);
      v2f b = *(const v2f*)(W3 + (16 * j + lm) * F1 + k0);
      acc = wmma_f32_k4(a, b, acc);
    }
    #pragma unroll
    for (int r = 0; r < 8; ++r)
      sfc3[wave][r + half * 8][16 * j + lm] = relu_f(acc[r]);
  }

  // ---- fc4: [16,64] x [64,64] -> output
  #pragma unroll
  for (int j = 0; j < 4; ++j) {
    float cb = b4[16 * j + lm];
    v8f acc = {cb, cb, cb, cb, cb, cb, cb, cb};
    #pragma unroll
    for (int kk = 0; kk < 16; ++kk) {
      int k0 = kk * 4 + half * 2;
      v2f a = *(const v2f*)&sfc3[wave][lm][k0];
      v2f b = *(const v2f*)(W4 + (16 * j + lm) * F2 + k0);
      acc = wmma_f32_k4(a, b, acc);
    }
    #pragma unroll
    for (int r = 0; r < 8; ++r)
      fc4_out[(size_t)(wc + r + half * 8) * FO + 16 * j + lm] = relu_f(acc[r]);
  }
}

// ---------------------------------------------------------------------------
// Kernel 3: fc5 head (67->3) per point + segment sums / counts (f32 atomics)
// ---------------------------------------------------------------------------
__global__ __launch_bounds__(256) void k_fc5_reduce(
    const float* __restrict__ points,      // [N,3]
    const int*   __restrict__ cluster,     // [N]
    const float* __restrict__ fc4,         // [C,64] (= d_out front)
    const float* __restrict__ W5,          // [3,67]
    const float* __restrict__ b5,          // [3]
    float* __restrict__ sums,              // [C,3], pre-zeroed
    float* __restrict__ cnt)               // [C],   pre-zeroed
{
  __shared__ float sW5[3 * (FO + 3)];
  for (int i = threadIdx.x; i < 3 * (FO + 3); i += 256) sW5[i] = W5[i];
  __syncthreads();

  const int p = blockIdx.x * 256 + threadIdx.x;
  const float x = points[p * 3 + 0];
  const float y = points[p * 3 + 1];
  const float z = points[p * 3 + 2];
  const int c = cluster[p];

  float a0 = b5[0] + x * sW5[0 * 67 + 0] + y * sW5[0 * 67 + 1] + z * sW5[0 * 67 + 2];
  float a1 = b5[1] + x * sW5[1 * 67 + 0] + y * sW5[1 * 67 + 1] + z * sW5[1 * 67 + 2];
  float a2 = b5[2] + x * sW5[2 * 67 + 0] + y * sW5[2 * 67 + 1] + z * sW5[2 * 67 + 2];

  const float4* f4 = (const float4*)(fc4 + (size_t)c * FO);  // 256B-aligned rows
  #pragma unroll
  for (int q = 0; q < FO / 4; ++q) {
    float4 f = f4[q];
    int k = q * 4;
    a0 += f.x * sW5[0 * 67 + 3 + k] + f.y * sW5[0 * 67 + 4 + k] +
          f.z * sW5[0 * 67 + 5 + k] + f.w * sW5[0 * 67 + 6 + k];
    a1 += f.x * sW5[1 * 67 + 3 + k] + f.y * sW5[1 * 67 + 4 + k] +
          f.z * sW5[1 * 67 + 5 + k] + f.w * sW5[1 * 67 + 6 + k];
    a2 += f.x * sW5[2 * 67 + 3 + k] + f.y * sW5[2 * 67 + 4 + k] +
          f.z * sW5[2 * 67 + 5 + k] + f.w * sW5[2 * 67 + 6 + k];
  }
  atomicAdd(&sums[(size_t)c * 3 + 0], relu_f(a0));
  atomicAdd(&sums[(size_t)c * 3 + 1], relu_f(a1));
  atomicAdd(&sums[(size_t)c * 3 + 2], relu_f(a2));
  atomicAdd(&cnt[c], 1.0f);
}

// ---------------------------------------------------------------------------
// Kernel 4: angles = sigmoid(sums / max(cnt, 1)) -> d_out[C*64 : C*64 + C*3]
// ---------------------------------------------------------------------------
__global__ __launch_bounds__(256) void k_angles(
    const float* __restrict__ sums, const float* __restrict__ cnt,
    float* __restrict__ out)
{
  int i = blockIdx.x * 256 + threadIdx.x;
  if (i >= NCLUST * 3) return;
  float m = sums[i] / fmaxf(cnt[i / 3], 1.0f);
  out[i] = 1.0f / (1.0f + __expf(-m));
}

// ---------------------------------------------------------------------------
extern "C" void kernel_launch(void* const* d_in, const int* in_sizes, int n_in,
                              void* d_out, int out_size, void* d_ws, size_t ws_size,
                              hipStream_t stream) {
  const float* points    = (const float*)d_in[0];
  const float* inv_feats = (const float*)d_in[1];
  const int*   cluster   = (const int*)  d_in[2];
  const float* W1 = (const float*)d_in[3];  const float* b1 = (const float*)d_in[4];
  const float* W2 = (const float*)d_in[5];  const float* b2 = (const float*)d_in[6];
  const float* W3 = (const float*)d_in[7];  const float* b3 = (const float*)d_in[8];
  const float* W4 = (const float*)d_in[9];  const float* b4 = (const float*)d_in[10];
  const float* W5 = (const float*)d_in[11]; const float* b5 = (const float*)d_in[12];

  float* out_fc4 = (float*)d_out;                      // [C,64]
  float* out_ang = out_fc4 + (size_t)NCLUST * FO;      // [C,3]

  float* mx   = (float*)d_ws;                          // [C,128]
  float* sums = mx + (size_t)NCLUST * F1;              // [C,3]
  float* cnt  = sums + (size_t)NCLUST * 3;             // [C]

  size_t zero_bytes = ((size_t)NCLUST * F1 + (size_t)NCLUST * 3 + NCLUST) * sizeof(float);
  hipMemsetAsync(d_ws, 0, zero_bytes, stream);

  k_point_mlp<<<NPTS / 128, 256, 0, stream>>>(inv_feats, cluster, W1, b1, W2, b2, mx);

  int cl_blocks = (NCLUST / 16 + 7) / 8;               // 391
  k_cluster_mlp<<<cl_blocks, 256, 0, stream>>>(mx, W3, b3, W4, b4, out_fc4);

  k_fc5_reduce<<<NPTS / 256, 256, 0, stream>>>(points, cluster, out_fc4, W5, b5, sums, cnt);

  int tot = NCLUST * 3;
  k_angles<<<(tot + 255) / 256, 256, 0, stream>>>(sums, cnt, out_ang);
}